// YoloLossLayer_33638183862875
// MI455X (gfx1250) — compile-verified
//
#include <hip/hip_runtime.h>
#include <math.h>
#include <stdint.h>

// Problem constants (must match reference)
#define B_    64
#define A_    5
#define NC_   80
#define GH_   52
#define GW_   52
#define HW_   (GH_ * GW_)          // 2704
#define MAXB_ 30
#define CH_   (NC_ + 5)            // 85

#define COORD_SCALE 5.0f
#define NOOBJ_SCALE 1.0f
#define OBJ_SCALE   5.0f
#define CLASS_SCALE 1.0f
#define OBJ_THRESH  0.5f
// NET_W/GW = 416/52 = 8
#define CELL_PIX    8.0f
#define INV_NETWH   (1.0f / (416.0f * 416.0f))

typedef float v8f __attribute__((ext_vector_type(8)));
typedef float v2f __attribute__((ext_vector_type(2)));

// Full-wave (32 lane) sum using one v_wmma_f32_16x16x4_f32.
// A = ones(16x4), B VGPR0 = VGPR1 = v  =>  for either legal B row->VGPR
// mapping, column n of B holds {v_n, v_n, v_{n+16}, v_{n+16}} (in some order),
// so D[m][n] = 2*(v_n + v_{n+16}) for every m. Element 0 of the C/D fragment
// in lane l is D[ l<16 ? 0 : 8 ][ l%16 ] = 2*(v_{l%16} + v_{l%16+16}).
// Halve (exact), then 4 xor-shuffle adds complete the 32-lane reduction
// (result broadcast to all lanes).
__device__ __forceinline__ float wave_sum_wmma(float v) {
    v2f a = {1.0f, 1.0f};
    v2f b = {v, v};
    v8f c = {};
    c = __builtin_amdgcn_wmma_f32_16x16x4_f32(
        /*neg_a=*/false, a, /*neg_b=*/false, b,
        /*c_mod=*/(short)0, c, /*reuse_a=*/false, /*reuse_b=*/false);
    float w = c[0] * 0.5f;               // v_{l%16} + v_{l%16+16}
    w += __shfl_xor(w, 1, 32);
    w += __shfl_xor(w, 2, 32);
    w += __shfl_xor(w, 4, 32);
    w += __shfl_xor(w, 8, 32);
    return w;
}

__global__ void zero_out_kernel(float* out) { out[0] = 0.0f; }

__global__ __launch_bounds__(256)
void yolo_loss_kernel(const float* __restrict__ net,   // (B, A, 85, HW) flat
                      const float* __restrict__ tgt,   // (B, A, 6, HW)
                      const float* __restrict__ amask, // (B, A, HW)
                      const float* __restrict__ tbox,  // (B, MAXB, 4)
                      const float* __restrict__ anch,  // (A, 2)
                      float* __restrict__ out)
{
    __shared__ float s_tb[MAXB_ * 4];
    __shared__ float s_wave[8];

    const int a   = blockIdx.y;
    const int b   = blockIdx.z;
    const int tid = threadIdx.x;
    const int s   = blockIdx.x * 256 + tid;

    // --- Stage the 30 true boxes (480 B) for this image into LDS using the
    // gfx1250 async LDS-DMA path: 60 lanes x GLOBAL_LOAD_ASYNC_TO_LDS_B64.
    // Generic LDS pointers carry the LDS offset in the low 32 bits
    if (tid < MAXB_ * 2) {
        const uint32_t lds_addr =
            (uint32_t)(uintptr_t)(&s_tb[0]) + (uint32_t)(tid * 8);
        const void* gaddr =
            (const char*)tbox + (size_t)b * (MAXB_ * 16) + (size_t)tid * 8;
        asm volatile("global_load_async_to_lds_b64 %0, %1, off"
                     :: "v"(lds_addr), "v"(gaddr)
                     : "memory");
    }
    // Issuing waves drain ASYNCcnt before the barrier; others pass instantly.
    asm volatile("s_wait_asynccnt 0x0" ::: "memory");
    __syncthreads();

    float loss = 0.0f;

    if (s < HW_) {
        const int    ba = b * A_ + a;
        const float* xp = net + (size_t)ba * CH_ * HW_ + s;   // channel stride HW_
        const float* tp = tgt + (size_t)ba * 6   * HW_ + s;
        const float  am = amask[(size_t)ba * HW_ + s];

        // Always-needed channels (focal + noobj path are unmasked).
        const float x0 = xp[0 * HW_];
        const float x1 = xp[1 * HW_];
        const float x2 = xp[2 * HW_];
        const float x3 = xp[3 * HW_];
        const float x4 = xp[4 * HW_];
        const float tconf = tp[4 * HW_];

        const float xy0   = 1.0f / (1.0f + expf(-x0));
        const float xy1   = 1.0f / (1.0f + expf(-x1));
        const float pconf = 1.0f / (1.0f + expf(-x4));

        // ---- Focal loss on confidence (needed for both obj and noobj) ----
        const float bce = -(tconf * logf(pconf) + (1.0f - tconf) * log1pf(-pconf));
        const float pt  = expf(-bce);
        const float om  = 1.0f - pt;
        const float fl  = om * om * bce;

        // ---- Decode predicted box, max IoU vs 30 true boxes ----
        const int   col = s % GW_;
        const int   row = s / GW_;
        const float aw  = anch[a * 2 + 0];
        const float ah  = anch[a * 2 + 1];
        const float px  = (xy0 + (float)col) * CELL_PIX;
        const float py  = (xy1 + (float)row) * CELL_PIX;
        const float pw  = expf(x2) * aw;
        const float ph  = expf(x3) * ah;
        const float bx1 = px - 0.5f * pw;
        const float by1 = py - 0.5f * ph;
        const float bx2 = bx1 + pw;
        const float by2 = by1 + ph;
        const float a1  = (bx2 - bx1) * (by2 - by1);

        float iou_max = 0.0f;
        #pragma unroll 6
        for (int j = 0; j < MAXB_; ++j) {
            const float t0 = s_tb[j * 4 + 0];
            const float t1 = s_tb[j * 4 + 1];
            const float t2 = s_tb[j * 4 + 2];
            const float t3 = s_tb[j * 4 + 3];
            const float wi = fminf(bx2, t2) - fmaxf(bx1, t0);
            const float hi = fminf(by2, t3) - fmaxf(by1, t1);
            const float inter = fmaxf(wi, 0.0f) * fmaxf(hi, 0.0f);
            const float a2 = (t2 - t0) * (t3 - t1);
            const float iou = inter / (a1 + a2 - inter + 1e-05f);
            iou_max = fmaxf(iou_max, iou);
        }
        const float noobj = (iou_max < OBJ_THRESH ? 1.0f : 0.0f) * (1.0f - am);

        loss = fl * (OBJ_SCALE * am + NOOBJ_SCALE * noobj);

        // ---- anchor_mask-gated terms: coords + class log-softmax ----
        // anchor_mask is ~2% dense, and every remaining term is multiplied by
        // it, so we skip the 80-channel class read entirely when am == 0.
        if (am != 0.0f) {
            const float t0 = tp[0 * HW_];
            const float t1 = tp[1 * HW_];
            const float t2 = tp[2 * HW_];
            const float t3 = tp[3 * HW_];
            const float t5 = tp[5 * HW_];

            const float ws  = 2.0f - t2 * t3 * INV_NETWH;
            const float dx0 = xy0 - t0, dx1 = xy1 - t1;
            const float dw0 = x2 - t2,  dw1 = x3 - t3;
            const float coords =
                (dx0 * dx0 + dx1 * dx1 + dw0 * dw0 + dw1 * dw1) * am * ws;

            const float* cp = xp + 5 * HW_;
            float m = -INFINITY;
            #pragma unroll 8
            for (int c = 0; c < NC_; ++c) m = fmaxf(m, cp[c * HW_]);
            float se = 0.0f;
            #pragma unroll 8
            for (int c = 0; c < NC_; ++c) se += expf(cp[c * HW_] - m);
            const int   tc = (int)t5;
            const float lt = cp[tc * HW_];
            const float clas = (logf(se) + m - lt) * am;

            loss += CLASS_SCALE * clas + COORD_SCALE * coords;
        }
    }

    // ---- block reduction: WMMA wave sum -> LDS -> atomicAdd ----
    // (all 256 threads reach here; EXEC is all-1s at the WMMA)
    const float wsum = wave_sum_wmma(loss);
    const int wid  = tid >> 5;
    const int lane = tid & 31;
    if (lane == 0) s_wave[wid] = wsum;
    __syncthreads();
    if (tid == 0) {
        float t = 0.0f;
        #pragma unroll
        for (int i = 0; i < 8; ++i) t += s_wave[i];
        atomicAdd(out, t * (1.0f / (float)B_));
    }
}

extern "C" void kernel_launch(void* const* d_in, const int* in_sizes, int n_in,
                              void* d_out, int out_size, void* d_ws, size_t ws_size,
                              hipStream_t stream) {
    const float* net = (const float*)d_in[0];  // net_out
    const float* tgt = (const float*)d_in[1];  // targets
    const float* am  = (const float*)d_in[2];  // anchor_mask
    const float* tb  = (const float*)d_in[3];  // true_bboxs
    const float* an  = (const float*)d_in[4];  // anchors
    float* out = (float*)d_out;

    zero_out_kernel<<<1, 1, 0, stream>>>(out);

    dim3 grid((HW_ + 255) / 256, A_, B_);      // (11, 5, 64)
    yolo_loss_kernel<<<grid, 256, 0, stream>>>(net, tgt, am, tb, an, out);
}